// NCSAMix_41910290874508
// MI455X (gfx1250) — compile-verified
//
#include <hip/hip_runtime.h>
#include <math.h>

typedef __attribute__((ext_vector_type(2))) float v2f;
typedef __attribute__((ext_vector_type(8))) float v8f;

#define INVT    14.285714285714286f   // 1/0.07
#define SHIFT   14.285714285714286f   // fixed softmax shift; valid since |logit| <= 1/T
#define KB      64
#define LSTR    68                    // padded LDS row stride (floats) -> conflict-free column reads
#define CSLICES 4
#define CN      128                   // columns per chunk (B panel rows)

__device__ __forceinline__ size_t anchor_off(int g, int B, int D) {
  // anchor row g -> features[(g % B), (g / B), :]
  return (size_t)(g % B) * (size_t)(2 * D) + (size_t)(g / B) * (size_t)D;
}

// ---------------------------------------------------------------------------
// Per-row auxiliary kernel: one workgroup per anchor row r.
//   pdotS[r]    = (anchor_r . points[label_r]) / T          (replaced diagonal)
//   samixval[r] = 0.5*(s-1)^2 with s from SLERP'd prototype (never materialized)
// ---------------------------------------------------------------------------
__global__ __launch_bounds__(256)
void ncsamix_rowaux(const float* __restrict__ feat, const float* __restrict__ samix,
                    const float* __restrict__ points, const int* __restrict__ labels,
                    const int* __restrict__ mixed, const float* __restrict__ lamb_p,
                    float* __restrict__ pdotS, float* __restrict__ samixval,
                    int B, int D) {
  const int r = blockIdx.x;
  const int tid = threadIdx.x;
  const float* arow = feat + anchor_off(r, B, D);
  const float* sx   = samix + (size_t)r * (size_t)D;
  const int li = labels[r % B];
  const int lj = labels[mixed[r] % B];
  const float* corr = points + (size_t)li * (size_t)D;
  const float* perm = points + (size_t)lj * (size_t)D;

  float ac = 0.f, cp = 0.f, sc = 0.f, sp = 0.f;
  for (int k = tid; k < D; k += 256) {
    const float a = arow[k], c = corr[k], p = perm[k], s = sx[k];
    ac = fmaf(a, c, ac); cp = fmaf(c, p, cp);
    sc = fmaf(s, c, sc); sp = fmaf(s, p, sp);
  }
  __shared__ float red[4][256];
  red[0][tid] = ac; red[1][tid] = cp; red[2][tid] = sc; red[3][tid] = sp;
  __syncthreads();
  for (int off = 128; off > 0; off >>= 1) {
    if (tid < off) {
      red[0][tid] += red[0][tid + off]; red[1][tid] += red[1][tid + off];
      red[2][tid] += red[2][tid + off]; red[3][tid] += red[3][tid + off];
    }
    __syncthreads();
  }
  if (tid == 0) {
    pdotS[r] = red[0][0] * INVT;
    const float ct = fminf(1.f, fmaxf(-1.f, red[1][0]));
    const float theta = acosf(ct);
    float s;
    if (theta < 1e-6f) {
      s = red[2][0];
    } else {
      const float st = sinf(theta);
      const float lam = *lamb_p;
      s = (sinf(lam * theta) / st) * red[2][0] + (sinf((1.f - lam) * theta) / st) * red[3][0];
    }
    const float d = s - 1.f;
    samixval[r] = 0.5f * d * d;
  }
}

// ---------------------------------------------------------------------------
// Row-sum accumulation helpers over one 16x16 WMMA C tile.
// C/D layout: lanes 0-15 hold M=e (VGPR e), lanes 16-31 hold M=8+e; N = lane&15.
// ---------------------------------------------------------------------------
__device__ __forceinline__ void acc_denom(const v8f c, int cb, int ml, int rhalf, int rb,
                                          int i0, int j0, bool isPrev, int colLimit,
                                          float* __restrict__ racc) {
#pragma unroll
  for (int e = 0; e < 8; ++e) {
    const int lr = rb + rhalf + e;
    const int lc = cb + ml;
    const bool valid = isPrev ? (lc < colLimit) : ((i0 + lr) != (j0 + lc));
    float contrib = valid ? __expf(c[e] * INVT - SHIFT) : 0.f;
#pragma unroll
    for (int off = 1; off < 16; off <<= 1) contrib += __shfl_xor(contrib, off, 32);
    if (ml == 0) atomicAdd(&racc[lr], contrib);
  }
}

__device__ __forceinline__ void acc_numer(const v8f c, int cb, int ml, int rhalf, int rb,
                                          int i0, int j0,
                                          const int* __restrict__ labR, const int* __restrict__ labC,
                                          const float* __restrict__ pdL, const float* __restrict__ ldL,
                                          float* __restrict__ rnum, float* __restrict__ rcnt) {
#pragma unroll
  for (int e = 0; e < 8; ++e) {
    const int lr = rb + rhalf + e;
    const int lc = cb + ml;
    float l = c[e] * INVT;
    if ((i0 + lr) == (j0 + lc)) l = pdL[lr];           // diagonal uses proto dot
    const float msk = (labR[lr] == labC[lc]) ? 1.f : 0.f;
    const float lp = (l - SHIFT) - ldL[lr];            // log_prob
    const float om = 1.f - __expf(lp);
    float contrib = msk * om * om * lp;                // focal-weighted positive term
    float cnt = msk;
#pragma unroll
    for (int off = 1; off < 16; off <<= 1) {
      contrib += __shfl_xor(contrib, off, 32);
      cnt     += __shfl_xor(cnt, off, 32);
    }
    if (ml == 0) { atomicAdd(&rnum[lr], contrib); atomicAdd(&rcnt[lr], cnt); }
  }
}

// ---------------------------------------------------------------------------
// WMMA Gram sweep. Block = 8 waves (256 thr) owns 64 rows, sweeps 128-col
// chunks (strided by blockIdx.y slice). Each wave computes a 32x32 macro-tile
// (2x2 of 16x16 -> 4 WMMAs per 4 LDS b64 loads). mode 0: denominator partials
// (N cols minus diagonal, plus NP prev-proto cols). mode 1: focal numerator +
// positive counts (N cols, diagonal = proto dot).
// ---------------------------------------------------------------------------
__global__ __launch_bounds__(256)
void ncsamix_gemm(const float* __restrict__ feat, const float* __restrict__ points,
                  const int* __restrict__ labels, const int* __restrict__ np_ptr,
                  const float* __restrict__ pdotS, const float* __restrict__ logD,
                  float* __restrict__ outA, float* __restrict__ outB,
                  int N, int B, int D, int mode) {
  __shared__ __align__(16) float As[64 * LSTR];
  __shared__ __align__(16) float Bs[CN * LSTR];
  __shared__ float racc[64];
  __shared__ float rcnt[64];
  __shared__ int   labR[64];
  __shared__ int   labC[CN];
  __shared__ float pdL[64];
  __shared__ float ldL[64];

  const int tid   = threadIdx.x;
  const int lane  = tid & 31;
  const int wave  = tid >> 5;
  const int i0    = blockIdx.x * 64;
  const int slice = blockIdx.y;

  if (tid < 64) {
    racc[tid] = 0.f;
    rcnt[tid] = 0.f;
    labR[tid] = labels[(i0 + tid) % B];
    if (mode == 1) { pdL[tid] = pdotS[i0 + tid]; ldL[tid] = logD[i0 + tid]; }
  }
  const int NP = *np_ptr;

  // 2x4 wave grid; each wave: 2 row strips x 2 col tiles (32x32 macro-tile)
  const int rb0   = (wave >> 2) * 32;
  const int rb1   = rb0 + 16;
  const int cb0   = (wave & 3) * 32;
  const int cb1   = cb0 + 16;
  const int ml    = lane & 15;
  const int kh    = (lane >> 4) * 2;       // A/B fragment K-half select
  const int rhalf = (lane >> 4) * 8;       // C fragment M-half select
  const int aoff0 = (rb0 + ml) * LSTR + kh;
  const int aoff1 = (rb1 + ml) * LSTR + kh;
  const int boff0 = (cb0 + ml) * LSTR + kh;
  const int boff1 = (cb1 + ml) * LSTR + kh;

  const int mainChunks  = N / CN;
  const int totalChunks = (mode == 0) ? mainChunks + (NP + CN - 1) / CN : mainChunks;

  for (int chunk = slice; chunk < totalChunks; chunk += CSLICES) {
    const bool isPrev   = (chunk >= mainChunks);
    const int  j0       = isPrev ? (chunk - mainChunks) * CN : chunk * CN;
    const int  colLimit = isPrev ? (NP - j0) : CN;
    __syncthreads();                       // previous chunk's LDS reads/atomics done
    if (mode == 1 && tid < CN) labC[tid] = labels[(j0 + tid) % B];

    v8f c00 = {}; v8f c01 = {}; v8f c10 = {}; v8f c11 = {};

    for (int kk = 0; kk < D; kk += KB) {
      __syncthreads();
      // Stage 64xKB panel of A-rows and CNxKB panel of B-rows (prefetch next K)
      for (int t = tid; t < (64 + CN) * 16; t += 256) {
        if (t < 64 * 16) {
          const int r  = t >> 4;
          const int cv = (t & 15) * 4;
          const float* src = feat + anchor_off(i0 + r, B, D) + kk + cv;
          *(float4*)(&As[r * LSTR + cv]) = *(const float4*)src;
          if (kk + KB < D) __builtin_prefetch(src + KB, 0, 0);
        } else {
          const int tt = t - 64 * 16;
          const int r  = tt >> 4;
          const int cv = (tt & 15) * 4;
          float4 vb;
          if (!isPrev) {
            const float* src = feat + anchor_off(j0 + r, B, D) + kk + cv;
            vb = *(const float4*)src;
            if (kk + KB < D) __builtin_prefetch(src + KB, 0, 0);
          } else if (r < colLimit) {
            const float* src = points + (size_t)(j0 + r) * (size_t)D + kk + cv;
            vb = *(const float4*)src;
            if (kk + KB < D) __builtin_prefetch(src + KB, 0, 0);
          } else {
            vb = make_float4(0.f, 0.f, 0.f, 0.f);
          }
          *(float4*)(&Bs[r * LSTR + cv]) = vb;
        }
      }
      __syncthreads();
#pragma unroll
      for (int k = 0; k < KB; k += 4) {
        const v2f a0 = *(const v2f*)(&As[aoff0 + k]);
        const v2f a1 = *(const v2f*)(&As[aoff1 + k]);
        const v2f b0 = *(const v2f*)(&Bs[boff0 + k]);
        const v2f b1 = *(const v2f*)(&Bs[boff1 + k]);
        c00 = __builtin_amdgcn_wmma_f32_16x16x4_f32(false, a0, false, b0, (short)0, c00, false, false);
        c01 = __builtin_amdgcn_wmma_f32_16x16x4_f32(false, a0, false, b1, (short)0, c01, false, false);
        c10 = __builtin_amdgcn_wmma_f32_16x16x4_f32(false, a1, false, b0, (short)0, c10, false, false);
        c11 = __builtin_amdgcn_wmma_f32_16x16x4_f32(false, a1, false, b1, (short)0, c11, false, false);
      }
    }

    if (mode == 0) {
      acc_denom(c00, cb0, ml, rhalf, rb0, i0, j0, isPrev, colLimit, racc);
      acc_denom(c01, cb1, ml, rhalf, rb0, i0, j0, isPrev, colLimit, racc);
      acc_denom(c10, cb0, ml, rhalf, rb1, i0, j0, isPrev, colLimit, racc);
      acc_denom(c11, cb1, ml, rhalf, rb1, i0, j0, isPrev, colLimit, racc);
    } else {
      acc_numer(c00, cb0, ml, rhalf, rb0, i0, j0, labR, labC, pdL, ldL, racc, rcnt);
      acc_numer(c01, cb1, ml, rhalf, rb0, i0, j0, labR, labC, pdL, ldL, racc, rcnt);
      acc_numer(c10, cb0, ml, rhalf, rb1, i0, j0, labR, labC, pdL, ldL, racc, rcnt);
      acc_numer(c11, cb1, ml, rhalf, rb1, i0, j0, labR, labC, pdL, ldL, racc, rcnt);
    }
  }

  __syncthreads();
  if (tid < 64) {
    outA[(size_t)slice * N + i0 + tid] = racc[tid];
    if (mode == 1) outB[(size_t)slice * N + i0 + tid] = rcnt[tid];
  }
}

// ---------------------------------------------------------------------------
// Combine slice partials of the denominator into logD (deterministic order).
// ---------------------------------------------------------------------------
__global__ void ncsamix_combine1(const float* __restrict__ part, float* __restrict__ logD, int N) {
  const int r = blockIdx.x * blockDim.x + threadIdx.x;
  if (r < N) {
    float s = 0.f;
#pragma unroll
    for (int sl = 0; sl < CSLICES; ++sl) s += part[(size_t)sl * N + r];
    logD[r] = logf(s);
  }
}

// ---------------------------------------------------------------------------
// Final scalar: loss_normal + loss_samix.
// ---------------------------------------------------------------------------
__global__ __launch_bounds__(256)
void ncsamix_finalize(const int* __restrict__ labels, const int* __restrict__ targets,
                      const float* __restrict__ numpart, const float* __restrict__ pospart,
                      const float* __restrict__ samixval, float* __restrict__ out,
                      int N, int B, int NT) {
  __shared__ float redA[256], redB[256];
  float sa = 0.f, sb = 0.f;
  for (int r = threadIdx.x; r < N; r += 256) {
    float num = 0.f, pos = 0.f;
#pragma unroll
    for (int sl = 0; sl < CSLICES; ++sl) {
      num += numpart[(size_t)sl * N + r];
      pos += pospart[(size_t)sl * N + r];
    }
    if (pos < 1e-6f) pos = 1.f;
    const float mlp = -(num / pos);        // -(T/base_T)=-1 folded in
    const int lab = labels[r % B];
    int curr = 0;
    for (int t = 0; t < NT; ++t) if (targets[t] == lab) { curr = 1; break; }
    sa += curr ? mlp : 0.f;
    sb += samixval[r];
  }
  redA[threadIdx.x] = sa; redB[threadIdx.x] = sb;
  __syncthreads();
  for (int off = 128; off > 0; off >>= 1) {
    if (threadIdx.x < off) {
      redA[threadIdx.x] += redA[threadIdx.x + off];
      redB[threadIdx.x] += redB[threadIdx.x + off];
    }
    __syncthreads();
  }
  if (threadIdx.x == 0) out[0] = (redA[0] + redB[0]) / (float)N;
}

// ---------------------------------------------------------------------------
extern "C" void kernel_launch(void* const* d_in, const int* in_sizes, int n_in,
                              void* d_out, int out_size, void* d_ws, size_t ws_size,
                              hipStream_t stream) {
  const float* feat    = (const float*)d_in[0];
  const float* samix   = (const float*)d_in[1];
  const float* points  = (const float*)d_in[2];
  const float* lamb    = (const float*)d_in[3];
  const int*   labels  = (const int*)d_in[4];
  const int*   targets = (const int*)d_in[5];
  const int*   mixed   = (const int*)d_in[6];
  const int*   npp     = (const int*)d_in[7];

  const int B  = in_sizes[4];          // 2048
  const int N  = 2 * B;                // 4096 anchor rows
  const int D  = in_sizes[1] / N;      // 1024
  const int NT = in_sizes[5];          // 900 target labels

  float* ws       = (float*)d_ws;      // ~240 KB total
  float* p1part   = ws;                                   // CSLICES*N
  float* logD     = p1part  + (size_t)CSLICES * N;        // N
  float* numpart  = logD    + N;                          // CSLICES*N
  float* pospart  = numpart + (size_t)CSLICES * N;        // CSLICES*N
  float* pdotS    = pospart + (size_t)CSLICES * N;        // N
  float* samixval = pdotS   + N;                          // N

  ncsamix_rowaux<<<N, 256, 0, stream>>>(feat, samix, points, labels, mixed, lamb,
                                        pdotS, samixval, B, D);
  dim3 g(N / 64, CSLICES);
  ncsamix_gemm<<<g, 256, 0, stream>>>(feat, points, labels, npp, pdotS, logD,
                                      p1part, pospart, N, B, D, /*mode=*/0);
  ncsamix_combine1<<<(N + 255) / 256, 256, 0, stream>>>(p1part, logD, N);
  ncsamix_gemm<<<g, 256, 0, stream>>>(feat, points, labels, npp, pdotS, logD,
                                      numpart, pospart, N, B, D, /*mode=*/1);
  ncsamix_finalize<<<1, 256, 0, stream>>>(labels, targets, numpart, pospart,
                                          samixval, (float*)d_out, N, B, NT);
}